// SparseWignerRotation_7232724927073
// MI455X (gfx1250) — compile-verified
//
#include <hip/hip_runtime.h>

// ---------------------------------------------------------------------------
// SparseWignerRotation for MI455X (gfx1250)
//   B=2048, C=128, irreps (l,mult) = (0..4, 8), TOTAL_DIM=200
//   out[b, block_l] = Zalpha * A_l * Zbeta * A_l^T * Zgamma * in[b, block_l]
// Strategy: fold the 5 small matrices into one dxd matrix R_l(b) per (b,l)
// (tiny setup kernels), then a single bandwidth-bound streaming pass applies
// R via f32 WMMA (v_wmma_f32_16x16x4_f32): M pads d<=9, K tiled by 4,
// N = 16-channel tiles. K/M padding of R is exact zeros, so out-of-range
// B rows need no masking (0 * finite = 0) -> branch-free inner loop.
// ---------------------------------------------------------------------------

typedef __attribute__((ext_vector_type(2))) float v2f;
typedef __attribute__((ext_vector_type(8))) float v8f;

#define N_BATCH   2048
#define N_CHAN    128
#define TOTAL_DIM 200
#define BATCH_STRIDE (TOTAL_DIM * N_CHAN)

__device__ __forceinline__ double factd(int n) {
  double r = 1.0;
  for (int i = 2; i <= n; ++i) r *= (double)i;
  return r;
}
__device__ __forceinline__ double ipow(double b, int e) {
  double r = 1.0;
  for (int i = 0; i < e; ++i) r *= b;
  return r;
}

// ---------------------------------------------------------------------------
// Kernel 1: compute constant A_l matrices (l = 1..4) on device in fp64,
// store zero-padded 16x16 fp32 row-major at wsA + (l-1)*256.
// A = Re( U * Dc(-pi/2, -pi/2, pi/2) * U^H )
// ---------------------------------------------------------------------------
__global__ void wigner_compute_A(float* __restrict__ wsA) {
  int t = threadIdx.x;
  if (t >= 4) return;
  const int l = t + 1;
  const int d = 2 * l + 1;
  const double PI = 3.14159265358979323846;

  // small Wigner d at beta = -pi/2
  double dm[9][9];
  const double cb2 = cos(-PI / 4.0);
  const double sb2 = sin(-PI / 4.0);
  for (int mp = -l; mp <= l; ++mp) {
    for (int m = -l; m <= l; ++m) {
      double pref = sqrt(factd(l + mp) * factd(l - mp) * factd(l + m) * factd(l - m));
      int s0 = (m - mp) > 0 ? (m - mp) : 0;
      int s1 = (l + m) < (l - mp) ? (l + m) : (l - mp);
      double tot = 0.0;
      for (int s = s0; s <= s1; ++s) {
        double den = factd(l + m - s) * factd(s) * factd(mp - m + s) * factd(l - mp - s);
        double sgn = ((mp - m + s) & 1) ? -1.0 : 1.0;
        tot += sgn / den * ipow(cb2, 2 * l + m - mp - 2 * s) * ipow(sb2, mp - m + 2 * s);
      }
      dm[mp + l][m + l] = pref * tot;
    }
  }

  // Dc[i][j] = exp(-i*mv_i*a) * dm * exp(-i*mv_j*g), a=-pi/2, g=pi/2
  double DcR[9][9], DcI[9][9];
  for (int i = 0; i < d; ++i) {
    for (int j = 0; j < d; ++j) {
      double a1 = (double)(i - l) * PI / 2.0;   // -mv_i * a
      double a2 = -(double)(j - l) * PI / 2.0;  // -mv_j * g
      double p1r = cos(a1), p1i = sin(a1);
      double p2r = cos(a2), p2i = sin(a2);
      DcR[i][j] = dm[i][j] * (p1r * p2r - p1i * p2i);
      DcI[i][j] = dm[i][j] * (p1r * p2i + p1i * p2r);
    }
  }

  // U matrix (complex)
  double UR[9][9] = {}, UI[9][9] = {};
  const double is2 = 1.0 / sqrt(2.0);
  UR[l][l] = 1.0;
  for (int m = 1; m <= l; ++m) {
    double pm = (m & 1) ? -1.0 : 1.0;  // (-1)^m
    UR[l + m][l + m] = pm * is2;
    UR[l + m][l - m] = is2;
    UI[l - m][l - m] = is2;
    UI[l - m][l + m] = -pm * is2;
  }

  // M1 = U * Dc
  double M1R[9][9], M1I[9][9];
  for (int i = 0; i < d; ++i) {
    for (int j = 0; j < d; ++j) {
      double sr = 0.0, si = 0.0;
      for (int k = 0; k < d; ++k) {
        sr += UR[i][k] * DcR[k][j] - UI[i][k] * DcI[k][j];
        si += UR[i][k] * DcI[k][j] + UI[i][k] * DcR[k][j];
      }
      M1R[i][j] = sr;
      M1I[i][j] = si;
    }
  }

  // A = Re(M1 * U^H);  (U^H)[k][j] = conj(U[j][k])
  float* Ap = wsA + (size_t)(l - 1) * 256;
  for (int i = 0; i < 16; ++i) {
    for (int j = 0; j < 16; ++j) {
      float v = 0.f;
      if (i < d && j < d) {
        double sr = 0.0;
        for (int k = 0; k < d; ++k)
          sr += M1R[i][k] * UR[j][k] + M1I[i][k] * UI[j][k];
        v = (float)sr;
      }
      Ap[i * 16 + j] = v;
    }
  }
}

// ---------------------------------------------------------------------------
// Kernel 2: one thread per (b, l): R = Zalpha * A * Zbeta * A^T * Zgamma,
// stored zero-padded 16x16 fp32 at wsR + (b*4 + l-1)*256.
// Z[i][i] = cos(|m_i| th); Z[i][d-1-i] += sign(m_i) sin(|m_i| th)
// ---------------------------------------------------------------------------
__global__ void wigner_compute_R(const float* __restrict__ wsA,
                                 const float* __restrict__ sc_a,
                                 const float* __restrict__ sc_b,
                                 const float* __restrict__ sc_g,
                                 float* __restrict__ wsR) {
  int idx = blockIdx.x * blockDim.x + threadIdx.x;
  if (idx >= N_BATCH * 4) return;
  const int b = idx >> 2;
  const int li = (idx & 3) + 1;
  const int d = 2 * li + 1;
  const float* A = wsA + (size_t)(li - 1) * 256;

  float ca[9], sa[9], cb[9], sb[9], cg[9], sg[9];
  for (int i = 0; i < d; ++i) {
    int mm = i - li;
    int am = mm < 0 ? -mm : mm;
    float sgn = mm < 0 ? -1.f : (mm > 0 ? 1.f : 0.f);
    int base = (b * 5 + am) * 2;                 // sincos layout (B, 5, {sin,cos})
    ca[i] = sc_a[base + 1]; sa[i] = sgn * sc_a[base + 0];
    cb[i] = sc_b[base + 1]; sb[i] = sgn * sc_b[base + 0];
    cg[i] = sc_g[base + 1]; sg[i] = sgn * sc_g[base + 0];
  }

  // T1 = A^T * Zgamma   (sparse Z: only cols j and d-1-j contribute)
  float T1[81], T2[81], T3[81];
  for (int i = 0; i < d; ++i)
    for (int j = 0; j < d; ++j)
      T1[i * 9 + j] = A[j * 16 + i] * cg[j] + A[(d - 1 - j) * 16 + i] * sg[d - 1 - j];
  // T2 = Zbeta * T1
  for (int i = 0; i < d; ++i)
    for (int j = 0; j < d; ++j)
      T2[i * 9 + j] = cb[i] * T1[i * 9 + j] + sb[i] * T1[(d - 1 - i) * 9 + j];
  // T3 = A * T2
  for (int i = 0; i < d; ++i)
    for (int j = 0; j < d; ++j) {
      float s = 0.f;
      for (int k = 0; k < d; ++k) s += A[i * 16 + k] * T2[k * 9 + j];
      T3[i * 9 + j] = s;
    }
  // R = Zalpha * T3, zero-padded store (padding zeros are load-bearing:
  // they annihilate unmasked out-of-range B rows in the apply kernel)
  float* Rp = wsR + (size_t)idx * 256;
  for (int i = 0; i < 16; ++i)
    for (int j = 0; j < 16; ++j) {
      float v = 0.f;
      if (i < d && j < d)
        v = ca[i] * T3[i * 9 + j] + sa[i] * T3[(d - 1 - i) * 9 + j];
      Rp[i * 16 + j] = v;
    }
}

// ---------------------------------------------------------------------------
// Kernel 3 (bandwidth path), templated on L so d/ksteps are constexpr and
// everything fully unrolls. 1 wave per (b, 16-channel tile); each wave loops
// the 8 mult slots: out_tile(16x16) = Rpad(16xK) @ in_tile(Kx16) via
// v_wmma_f32_16x16x4_f32.
// A fragment layout (ISA 32-bit A 16x4): lanes 0-15 hold M=lane, K={0,1};
// lanes 16-31 hold M=lane-16, K={2,3}. C/D: VGPR v -> row v + 8*(lane>=16).
// B rows >= d are loaded with the row index clamped to d-1 (in-bounds) and
// NOT masked: the matching R columns are exact zeros.
// ---------------------------------------------------------------------------
template <int L>
__global__ __launch_bounds__(256) void wigner_apply_l(const float* __restrict__ in,
                                                      const float* __restrict__ wsR,
                                                      float* __restrict__ out) {
  constexpr int D = 2 * L + 1;
  constexpr int KSTEPS = (D + 3) / 4;

  const int lane = threadIdx.x & 31;
  const int wave = threadIdx.x >> 5;   // 0..7 -> channel tile
  const int b = blockIdx.x;

  const int M = lane & 15;
  const int half = lane >> 4;

  // A fragments: register-resident for the whole block
  const float* Rp = wsR + ((size_t)b * 4 + (L - 1)) * 256;
  v2f afrag[KSTEPS];
#pragma unroll
  for (int kk = 0; kk < KSTEPS; ++kk) {
    const int k0 = kk * 4 + half * 2;
    afrag[kk].x = Rp[M * 16 + k0];
    afrag[kk].y = Rp[M * 16 + k0 + 1];
  }

  const int col = wave * 16 + (lane & 15);
  const size_t blockOff = (size_t)b * BATCH_STRIDE + (size_t)(8 * L * L) * N_CHAN + col;
  const float* inB = in + blockOff;
  float* outB = out + blockOff;

  // Per-lane row base pointers (row clamped to D-1 keeps addresses in-bounds;
  // R's zero K-columns make the loaded value irrelevant). Computed once, so
  // the unrolled loop is global_load_b32 with immediate offsets.
  const float* ldp[KSTEPS][2];
#pragma unroll
  for (int kk = 0; kk < KSTEPS; ++kk) {
    const int j0 = kk * 4 + half * 2;
    const int r0 = (j0 < D) ? j0 : (D - 1);
    const int r1 = (j0 + 1 < D) ? (j0 + 1) : (D - 1);
    ldp[kk][0] = inB + (size_t)r0 * N_CHAN;
    ldp[kk][1] = inB + (size_t)r1 * N_CHAN;
  }
  float* stp = outB + (size_t)(half * 8) * N_CHAN;  // row base for this half

#pragma unroll
  for (int m = 0; m < 8; ++m) {
    const int rowoff = m * D * N_CHAN;  // fits the 24-bit immediate offset
    if (m < 7)
      __builtin_prefetch(inB + rowoff + D * N_CHAN, 0, 1);

    v8f acc = {};
#pragma unroll
    for (int kk = 0; kk < KSTEPS; ++kk) {
      v2f bfrag;
      bfrag.x = ldp[kk][0][rowoff];
      bfrag.y = ldp[kk][1][rowoff];
      acc = __builtin_amdgcn_wmma_f32_16x16x4_f32(
          /*neg_a=*/false, afrag[kk], /*neg_b=*/false, bfrag,
          /*c_mod=*/(short)0, acc, /*reuse_a=*/false, /*reuse_b=*/false);
    }
    // Store valid rows: condition is constexpr-per-v relative to `half`
    // (true / lanes 0-15 only / dead), so no branches, just a lane mask.
#pragma unroll
    for (int v = 0; v < 8; ++v) {
      const bool ok0 = (v < D);       // half == 0 row = v
      const bool ok1 = (v + 8 < D);   // half == 1 row = v + 8
      if (ok0 && ok1) {
        stp[rowoff + v * N_CHAN] = acc[v];
      } else if (ok0) {
        if (half == 0) stp[rowoff + v * N_CHAN] = acc[v];
      }
      // !ok0 && !ok1: dead code (never for v < D <= 9 with half==0... v>=D)
    }
  }
}

// l=0 block is the identity: straight float4 copy of rows 0..7.
__global__ void wigner_copy_l0(const float* __restrict__ in, float* __restrict__ out) {
  int tid = blockIdx.x * blockDim.x + threadIdx.x;   // N_BATCH * 256 threads
  int b = tid >> 8;
  int rem = (tid & 255) << 2;                        // 8 rows * 128 ch = 1024 floats
  size_t off = (size_t)b * BATCH_STRIDE + rem;
  const float4* s = (const float4*)(in + off);
  float4* p = (float4*)(out + off);
  *p = *s;
}

extern "C" void kernel_launch(void* const* d_in, const int* in_sizes, int n_in,
                              void* d_out, int out_size, void* d_ws, size_t ws_size,
                              hipStream_t stream) {
  (void)in_sizes; (void)n_in; (void)out_size; (void)ws_size;
  const float* in   = (const float*)d_in[0];
  const float* sc_a = (const float*)d_in[1];
  const float* sc_b = (const float*)d_in[2];
  const float* sc_g = (const float*)d_in[3];
  float* out = (float*)d_out;

  float* wsA = (float*)d_ws;          // 4 * 256 floats
  float* wsR = wsA + 4 * 256;         // 2048 * 4 * 256 floats (~8 MB)

  hipLaunchKernelGGL(wigner_compute_A, dim3(1), dim3(32), 0, stream, wsA);
  hipLaunchKernelGGL(wigner_compute_R, dim3((N_BATCH * 4 + 255) / 256), dim3(256), 0, stream,
                     wsA, sc_a, sc_b, sc_g, wsR);
  hipLaunchKernelGGL(wigner_copy_l0, dim3(N_BATCH), dim3(256), 0, stream, in, out);
  hipLaunchKernelGGL(wigner_apply_l<1>, dim3(N_BATCH), dim3(256), 0, stream, in, wsR, out);
  hipLaunchKernelGGL(wigner_apply_l<2>, dim3(N_BATCH), dim3(256), 0, stream, in, wsR, out);
  hipLaunchKernelGGL(wigner_apply_l<3>, dim3(N_BATCH), dim3(256), 0, stream, in, wsR, out);
  hipLaunchKernelGGL(wigner_apply_l<4>, dim3(N_BATCH), dim3(256), 0, stream, in, wsR, out);
}